// LDPCDecoder_42064909697029
// MI455X (gfx1250) — compile-verified
//
#include <hip/hip_runtime.h>
#include <stdint.h>

// Problem constants (from reference setup_inputs)
#define N_VAR   16384
#define M_CHK   8192
#define K_MSG   8192
#define DV      3
#define BATCH   512
#define N_EDGE  (N_VAR * DV)        // 49152
#define DC      6                   // every check has exactly 6 edges
#define MAX_IT  10
#define SCALE_LC 200.0f             // 2 / (10^(-20/10))

// ---------------- CDNA5 async global->LDS path (guarded, probe-verified) ----
#if defined(__has_builtin)
# if __has_builtin(__builtin_amdgcn_global_load_async_to_lds_b32) && \
     __has_builtin(__builtin_amdgcn_s_wait_asynccnt)
#  define HAVE_ASYNC_LDS 1
# endif
#endif
#ifndef HAVE_ASYNC_LDS
# define HAVE_ASYNC_LDS 0
#endif

#if HAVE_ASYNC_LDS
typedef __attribute__((address_space(3))) int* lds_i32_ptr;
typedef __attribute__((address_space(1))) int* glb_i32_ptr;
__device__ __forceinline__ void async_ld_f32(void* lds_generic, const void* gaddr) {
  // generic shared addr low 32 bits == LDS offset (flat aperture rule)
  lds_i32_ptr l = (lds_i32_ptr)(uint32_t)(uintptr_t)lds_generic;
  glb_i32_ptr g = (glb_i32_ptr)(uintptr_t)gaddr;
  __builtin_amdgcn_global_load_async_to_lds_b32(g, l, 0, 0);
}
#endif

// ---------------- fast hardware transcendentals ----------------
#if defined(__has_builtin)
# if __has_builtin(__builtin_amdgcn_tanhf)
#  define FAST_TANH(x) __builtin_amdgcn_tanhf(x)
# endif
#endif
#ifndef FAST_TANH
__device__ __forceinline__ float fast_tanh_impl(float x) {
  float e = __expf(2.0f * x);               // v_exp_f32
  return (e - 1.0f) * __frcp_rn(e + 1.0f);
}
# define FAST_TANH(x) fast_tanh_impl(x)
#endif

// ---------------- check-node pass (the only gathering pass) ----------------
// One block handles CPB checks for the whole batch (2 f32 per thread),
// double-buffering the next check's 6 Lq rows into LDS with async loads.
// Each wave reads back exactly the LDS lanes it issued, and async loads
// complete in order per wave, so s_wait_asynccnt alone fences the buffer
// (no block barriers needed).
#define CPB 8
__global__ __launch_bounds__(256) void k_check(const float* __restrict__ Lq,
                                               float* __restrict__ Lr,
                                               const int* __restrict__ chk) {
  const int tid = threadIdx.x;
  const int b2  = tid * 2;                 // batch pair [b2, b2+1]
  const int c0  = blockIdx.x * CPB;

#if HAVE_ASYNC_LDS
  __shared__ float buf[2][DC][BATCH];      // 24 KB
#pragma unroll
  for (int j = 0; j < DC; ++j) {
    const float* src = Lq + (size_t)chk[c0 * DC + j] * BATCH + b2;
    async_ld_f32(&buf[0][j][b2],     src);
    async_ld_f32(&buf[0][j][b2 + 1], src + 1);
  }
#endif

  for (int i = 0; i < CPB; ++i) {
    const int c = c0 + i;
#if HAVE_ASYNC_LDS
    if (i + 1 < CPB) {
#pragma unroll
      for (int j = 0; j < DC; ++j) {
        const float* src = Lq + (size_t)chk[(c + 1) * DC + j] * BATCH + b2;
        async_ld_f32(&buf[(i + 1) & 1][j][b2],     src);
        async_ld_f32(&buf[(i + 1) & 1][j][b2 + 1], src + 1);
      }
      __builtin_amdgcn_s_wait_asynccnt(12);   // current buffer's 12 loads done
    } else {
      __builtin_amdgcn_s_wait_asynccnt(0);
    }
#endif
    int e[DC];
#pragma unroll
    for (int j = 0; j < DC; ++j) e[j] = chk[c * DC + j];

    float lm0[DC], lm1[DC];
    int ng0[DC], ng1[DC];
    float slog0 = 0.0f, slog1 = 0.0f;
    int par0 = 0, par1 = 0;
#pragma unroll
    for (int j = 0; j < DC; ++j) {
#if HAVE_ASYNC_LDS
      float2 q = *(const float2*)&buf[i & 1][j][b2];
#else
      float2 q = *(const float2*)(Lq + (size_t)e[j] * BATCH + b2);
#endif
      float t0 = FAST_TANH(0.5f * q.x);
      float t1 = FAST_TANH(0.5f * q.y);
      lm0[j] = __logf(fabsf(t0) + 1e-12f);
      lm1[j] = __logf(fabsf(t1) + 1e-12f);
      ng0[j] = (t0 < 0.0f) ? 1 : 0;
      ng1[j] = (t1 < 0.0f) ? 1 : 0;
      slog0 += lm0[j]; par0 ^= ng0[j];
      slog1 += lm1[j]; par1 ^= ng1[j];
    }
#pragma unroll
    for (int j = 0; j < DC; ++j) {
      float X0 = ((par0 ^ ng0[j]) ? -1.0f : 1.0f) * __expf(slog0 - lm0[j]);
      float X1 = ((par1 ^ ng1[j]) ? -1.0f : 1.0f) * __expf(slog1 - lm1[j]);
      X0 = fminf(fmaxf(X0, -1.0f + 1e-7f), 1.0f - 1e-7f);
      X1 = fminf(fmaxf(X1, -1.0f + 1e-7f), 1.0f - 1e-7f);
      float2 r;
      r.x = __logf(1.0f + X0) - __logf(1.0f - X0);   // 2*atanh(X0)
      r.y = __logf(1.0f + X1) - __logf(1.0f - X1);
      *(float2*)(Lr + (size_t)e[j] * BATCH + b2) = r;
    }
  }
}

// ---------------- variable-node pass (pure contiguous, float4) -------------
__global__ __launch_bounds__(256) void k_var(const float* __restrict__ LcT,
                                             const float* __restrict__ Lr,
                                             float* __restrict__ Lq) {
  const size_t NB = (size_t)N_VAR * BATCH;
  size_t t = ((size_t)blockIdx.x * 256 + threadIdx.x) * 4;
  float4 lc = *(const float4*)(LcT + t);
  float4 r0 = *(const float4*)(Lr + t);
  float4 r1 = *(const float4*)(Lr + t + NB);
  float4 r2 = *(const float4*)(Lr + t + 2 * NB);
  float4 s;
  s.x = lc.x + r0.x + r1.x + r2.x;
  s.y = lc.y + r0.y + r1.y + r2.y;
  s.z = lc.z + r0.z + r1.z + r2.z;
  s.w = lc.w + r0.w + r1.w + r2.w;
  float4 o;
  o.x = s.x - r0.x; o.y = s.y - r0.y; o.z = s.z - r0.z; o.w = s.w - r0.w;
  *(float4*)(Lq + t) = o;
  o.x = s.x - r1.x; o.y = s.y - r1.y; o.z = s.z - r1.z; o.w = s.w - r1.w;
  *(float4*)(Lq + t + NB) = o;
  o.x = s.x - r2.x; o.y = s.y - r2.y; o.z = s.z - r2.z; o.w = s.w - r2.w;
  *(float4*)(Lq + t + 2 * NB) = o;
}

// ---------------- setup kernels ----------------
__global__ __launch_bounds__(256) void k_zero_i32(int* p, int n) {
  int i = blockIdx.x * 256 + threadIdx.x;
  if (i < n) p[i] = 0;
}

__global__ __launch_bounds__(256) void k_build(const int* __restrict__ echk,
                                               int* __restrict__ chk,
                                               int* __restrict__ cnt) {
  int e = blockIdx.x * 256 + threadIdx.x;
  if (e < N_EDGE) {
    int c = echk[e];
    int p = atomicAdd(&cnt[c], 1);
    chk[c * DC + p] = e;
  }
}

// sort each check's 6 edge ids ascending -> deterministic order across replays
__global__ __launch_bounds__(256) void k_sortchk(int* __restrict__ chk) {
  int c = blockIdx.x * 256 + threadIdx.x;
  if (c >= M_CHK) return;
  int a[DC];
#pragma unroll
  for (int j = 0; j < DC; ++j) a[j] = chk[c * DC + j];
#pragma unroll
  for (int i = 1; i < DC; ++i) {
    int key = a[i], j = i - 1;
    while (j >= 0 && a[j] > key) { a[j + 1] = a[j]; --j; }
    a[j + 1] = key;
  }
#pragma unroll
  for (int j = 0; j < DC; ++j) chk[c * DC + j] = a[j];
}

// LcT[v][b] = 200 * y[b][v]  (tiled transpose, both sides coalesced)
__global__ __launch_bounds__(256) void k_lct(const float* __restrict__ y,
                                             float* __restrict__ LcT) {
  __shared__ float tile[32][33];
  const int sv = blockIdx.x * 32;      // variable dim
  const int sb = blockIdx.y * 32;      // batch dim
#pragma unroll
  for (int r = 0; r < 4; ++r) {
    int b = sb + threadIdx.y + r * 8;
    int v = sv + threadIdx.x;
    tile[threadIdx.y + r * 8][threadIdx.x] = y[(size_t)b * N_VAR + v];
  }
  __syncthreads();
#pragma unroll
  for (int r = 0; r < 4; ++r) {
    int v = sv + threadIdx.y + r * 8;
    int b = sb + threadIdx.x;
    LcT[(size_t)v * BATCH + b] = SCALE_LC * tile[threadIdx.x][threadIdx.y + r * 8];
  }
}

// Lq init: edges of v are {v, v+n, v+2n}; float4 wide
__global__ __launch_bounds__(256) void k_init(const float* __restrict__ LcT,
                                              float* __restrict__ Lq) {
  const size_t NB = (size_t)N_VAR * BATCH;
  size_t t = ((size_t)blockIdx.x * 256 + threadIdx.x) * 4;
  float4 lc = *(const float4*)(LcT + t);
  *(float4*)(Lq + t)          = lc;
  *(float4*)(Lq + t + NB)     = lc;
  *(float4*)(Lq + t + 2 * NB) = lc;
}

// ---------------- posterior + hard decision, transposed out ----------------
__global__ __launch_bounds__(256) void k_out(const float* __restrict__ LcT,
                                             const float* __restrict__ Lr,
                                             float* __restrict__ out) {
  __shared__ float tile[32][33];
  const size_t NB = (size_t)N_VAR * BATCH;
  const int sv = blockIdx.x * 32;      // over K_MSG
  const int sb = blockIdx.y * 32;
#pragma unroll
  for (int r = 0; r < 4; ++r) {
    int v = sv + threadIdx.y + r * 8;
    int b = sb + threadIdx.x;
    size_t t = (size_t)v * BATCH + b;
    tile[threadIdx.y + r * 8][threadIdx.x] =
        LcT[t] + Lr[t] + Lr[t + NB] + Lr[t + 2 * NB];
  }
  __syncthreads();
#pragma unroll
  for (int r = 0; r < 4; ++r) {
    int b = sb + threadIdx.y + r * 8;
    int v = sv + threadIdx.x;
    float L = tile[threadIdx.x][threadIdx.y + r * 8];
    out[(size_t)b * K_MSG + v] = (L <= 0.0f) ? 1.0f : 0.0f;
  }
}

// ---------------- launcher ----------------
extern "C" void kernel_launch(void* const* d_in, const int* in_sizes, int n_in,
                              void* d_out, int out_size, void* d_ws, size_t ws_size,
                              hipStream_t stream) {
  (void)in_sizes; (void)n_in; (void)out_size; (void)ws_size;
  const float* y    = (const float*)d_in[0];
  const int*   echk = (const int*)d_in[2];
  float* out = (float*)d_out;

  char* ws = (char*)d_ws;
  float* LcT = (float*)ws;                                  // N*B
  float* Lq  = LcT + (size_t)N_VAR * BATCH;                 // E*B
  float* Lr  = Lq  + (size_t)N_EDGE * BATCH;                // E*B
  int*   chk = (int*)(Lr + (size_t)N_EDGE * BATCH);         // M*6
  int*   cnt = chk + M_CHK * DC;                            // M

  // deterministic check-adjacency build (each check has exactly 6 edges)
  k_zero_i32<<<(M_CHK + 255) / 256, 256, 0, stream>>>(cnt, M_CHK);
  k_build<<<(N_EDGE + 255) / 256, 256, 0, stream>>>(echk, chk, cnt);
  k_sortchk<<<(M_CHK + 255) / 256, 256, 0, stream>>>(chk);

  // channel LLRs, transposed + scaled; Lq init
  k_lct<<<dim3(N_VAR / 32, BATCH / 32), dim3(32, 8), 0, stream>>>(y, LcT);
  k_init<<<(N_VAR * BATCH) / (256 * 4), 256, 0, stream>>>(LcT, Lq);

  const int check_grid = M_CHK / CPB;                       // 1024
  const int var_grid   = (N_VAR * BATCH) / (256 * 4);       // 8192
  for (int it = 0; it < MAX_IT; ++it) {
    k_check<<<check_grid, 256, 0, stream>>>(Lq, Lr, chk);
    k_var<<<var_grid, 256, 0, stream>>>(LcT, Lr, Lq);
  }

  k_out<<<dim3(K_MSG / 32, BATCH / 32), dim3(32, 8), 0, stream>>>(LcT, Lr, out);
}